// SPGAT_29918742184373
// MI455X (gfx1250) — compile-verified
//
#include <hip/hip_runtime.h>
#include <math.h>

#define NEG_SLOPE 0.2f

typedef __attribute__((ext_vector_type(2))) float v2f;
typedef __attribute__((ext_vector_type(8))) float v8f;

// ---------------------------------------------------------------------------
// C = A[M,K] @ B[K,N] (+ bias[N]), row-major, via V_WMMA_F32_16X16X4_F32.
// 128 threads = 4 waves; each wave owns one 16x16 output tile.
// Requires M%16==0, N%64==0, K%4==0 (true for all call sites). No divergence,
// EXEC stays all-ones around the WMMA as the ISA requires.
// ---------------------------------------------------------------------------
__global__ __launch_bounds__(128) void spgat_gemm_wmma(
    const float* __restrict__ A, const float* __restrict__ B,
    const float* __restrict__ bias, float* __restrict__ C,
    int M, int N, int K)
{
  const int lane = threadIdx.x & 31;
  const int wave = threadIdx.x >> 5;
  const int half = lane >> 4;      // 0: lanes 0-15, 1: lanes 16-31
  const int ln   = lane & 15;
  const int mt   = blockIdx.x;
  const int nt   = blockIdx.y * 4 + wave;
  const int row  = mt * 16 + ln;   // A-fragment row for this lane
  const int col  = nt * 16 + ln;   // B/C-fragment column for this lane

  v8f acc = {};
  const float* Arow = A + (size_t)row * K;
  for (int k = 0; k < K; k += 4) {
    // A 16x4 f32 frag: V0 = K{0 | 2}, V1 = K{1 | 3}  (ISA 7.12.2)
    v2f a = *(const v2f*)(Arow + k + 2 * half);
    // B 4x16 f32 frag: symmetric, N striped across lanes
    v2f b;
    b.x = B[(size_t)(k + 2 * half)     * N + col];
    b.y = B[(size_t)(k + 2 * half + 1) * N + col];
    acc = __builtin_amdgcn_wmma_f32_16x16x4_f32(
        /*neg_a=*/false, a, /*neg_b=*/false, b,
        /*c_mod=*/(short)0, acc, /*reuse_a=*/false, /*reuse_b=*/false);
  }
  const float bv = bias ? bias[col] : 0.0f;
#pragma unroll
  for (int r = 0; r < 8; ++r) {     // C/D: VGPR r -> row (r + 8*half)
    C[(size_t)(mt * 16 + half * 8 + r) * N + col] = acc[r] + bv;
  }
}

__global__ void spgat_fill(float* __restrict__ p, float v, long long n) {
  long long i = (long long)blockIdx.x * blockDim.x + threadIdx.x;
  if (i < n) p[i] = v;
}

__global__ void spgat_ifill(int* __restrict__ p, int v, long long n) {
  long long i = (long long)blockIdx.x * blockDim.x + threadIdx.x;
  if (i < n) p[i] = v;
}

// per (node,head): attention score dot-products against a_src / a_dst
__global__ void spgat_scores(const float* __restrict__ hW,
                             const float* __restrict__ asrc,
                             const float* __restrict__ adst,
                             float* __restrict__ ss, float* __restrict__ sd,
                             int Nn) {
  int i = blockIdx.x * blockDim.x + threadIdx.x;   // i = node*4 + head
  if (i >= Nn * 4) return;
  int hd = i & 3;
  const float* hp  = hW + (size_t)i * 64;          // hW[(n*4+hd)*64]
  const float* as_ = asrc + hd * 64;
  const float* ad_ = adst + hd * 64;
  float s1 = 0.f, s2 = 0.f;
#pragma unroll 8
  for (int c = 0; c < 64; ++c) { float v = hp[c]; s1 += v * as_[c]; s2 += v * ad_[c]; }
  ss[i] = s1; sd[i] = s2;
}

// ---------------- CSR build (once; reused by all 3 layers) ----------------
// logical edge list = [E real edges] ++ [Nn self loops]

__global__ void spgat_count(const int* __restrict__ ei, int E, int Nn,
                            int* __restrict__ cnt) {
  long long i = (long long)blockIdx.x * blockDim.x + threadIdx.x;
  if (i >= (long long)E + Nn) return;
  int d = (i < E) ? ei[(long long)E + i] : (int)(i - E);
  atomicAdd(&cnt[d], 1);
}

// single-block hierarchical exclusive scan of cnt[0..Nn) -> start, cursor
__global__ __launch_bounds__(1024) void spgat_scan(
    const int* __restrict__ cnt, int* __restrict__ start,
    int* __restrict__ cursor, int Nn, int total) {
  __shared__ int s[1024];
  const int t = threadIdx.x;
  const int per = (Nn + 1023) / 1024;
  const int b0 = t * per;
  const int e0 = (b0 + per < Nn) ? (b0 + per) : Nn;
  int sum = 0;
  for (int i = b0; i < e0; ++i) sum += cnt[i];
  s[t] = sum;
  __syncthreads();
  for (int off = 1; off < 1024; off <<= 1) {   // Hillis-Steele inclusive scan
    int v = (t >= off) ? s[t - off] : 0;
    __syncthreads();
    s[t] += v;
    __syncthreads();
  }
  int run = (t == 0) ? 0 : s[t - 1];           // exclusive base for this range
  for (int i = b0; i < e0; ++i) { start[i] = run; cursor[i] = run; run += cnt[i]; }
  if (t == 0) start[Nn] = total;
}

__global__ void spgat_scatter(const int* __restrict__ ei, int E, int Nn,
                              int* __restrict__ cursor, int* __restrict__ srcl) {
  long long i = (long long)blockIdx.x * blockDim.x + threadIdx.x;
  if (i >= (long long)E + Nn) return;
  int s, d;
  if (i < E) { s = ei[i]; d = ei[(long long)E + i]; }
  else       { s = d = (int)(i - E); }
  int pos = atomicAdd(&cursor[d], 1);
  srcl[pos] = s;
}

// ---------------------------------------------------------------------------
// Fused per-dst softmax aggregation. One 256-thread block per dst node:
//   phase A: LDS tree-reduce segment max per head (no CAS loops)
//   phase B: chunked over incoming edges; softmax weights staged in LDS;
//            feature gather hW[src] is a contiguous 1 KB row per edge read
//            by all 256 lanes -> perfectly coalesced streaming, zero atomics.
// mode 0: concat heads + bias + ELU -> out[N,256]
// mode 1: mean over heads + bias    -> out[N,64]
// ---------------------------------------------------------------------------
__global__ __launch_bounds__(256) void spgat_gather(
    const int* __restrict__ srcl, const int* __restrict__ start,
    const float* __restrict__ ss, const float* __restrict__ sd,
    const float* __restrict__ hW, const float* __restrict__ bias,
    float* __restrict__ out, int mode) {
  __shared__ int   s_src[64];
  __shared__ float s_ex[256];
  __shared__ float s_red[256];
  __shared__ float s_m[4];

  const int n   = blockIdx.x;
  const int t   = threadIdx.x;
  const int base = start[n];
  const int deg  = start[n + 1] - base;   // >= 1 (self loop)

  // ---- phase A: segment max per head ----
  {
    const int hd = t & 3;
    const int j0 = t >> 2;                 // 64 threads per head
    const float sdn = sd[n * 4 + hd];
    float lm = -3.4e38f;
    for (int j = j0; j < deg; j += 64) {
      float v = ss[srcl[base + j] * 4 + hd] + sdn;
      v = v > 0.f ? v : NEG_SLOPE * v;
      lm = lm > v ? lm : v;
    }
    s_red[t] = lm;                         // layout: t = j0*4 + hd
    __syncthreads();
    for (int off = 128; off >= 4; off >>= 1) {   // offs all %4==0 -> per-head
      if (t < off) { float o = s_red[t + off]; if (o > s_red[t]) s_red[t] = o; }
      __syncthreads();
    }
    if (t < 4) s_m[t] = s_red[t];
    __syncthreads();
  }

  // ---- phase B: denom + weighted feature accumulation ----
  const int hd = t >> 6;                   // head for accumulation role
  const int c  = t & 63;                   // channel (also edge slot for ex)
  const float sdn = sd[n * 4 + hd];
  const float mh  = s_m[hd];
  float dpart = 0.f;                       // this thread's partial denom
  float accv  = 0.f;                       // unnormalized sum for (hd,c)

  const int nchunks = (deg + 63) / 64;
  for (int ch = 0; ch < nchunks; ++ch) {
    const int cbase = ch * 64;
    const int clen  = (deg - cbase < 64) ? (deg - cbase) : 64;
    if (t < clen) s_src[t] = srcl[base + cbase + t];
    __syncthreads();
    if (c < clen) {                        // thread (hd, j=c) computes ex
      float v = ss[s_src[c] * 4 + hd] + sdn;
      v = v > 0.f ? v : NEG_SLOPE * v;
      float ex = __expf(v - mh);
      s_ex[hd * 64 + c] = ex;
      dpart += ex;
    }
    __syncthreads();
    for (int j = 0; j < clen; ++j)         // coalesced 1 KB row per edge
      accv += s_ex[hd * 64 + j] * hW[(size_t)s_src[j] * 256 + t];
    __syncthreads();
  }

  // reduce denom within each head's 64-thread group
  s_red[t] = dpart;
  __syncthreads();
  for (int off = 32; off >= 1; off >>= 1) {
    if (c < off) s_red[hd * 64 + c] += s_red[hd * 64 + c + off];
    __syncthreads();
  }
  const float val = accv / s_red[hd * 64];

  if (mode == 0) {                         // concat + bias + ELU
    float v = val + bias[t];
    out[(size_t)n * 256 + t] = v > 0.f ? v : (__expf(v) - 1.f);
  } else {                                 // mean over heads + bias
    s_ex[t] = val;
    __syncthreads();
    if (t < 64) {
      float s = 0.25f * (s_ex[t] + s_ex[64 + t] + s_ex[128 + t] + s_ex[192 + t]);
      out[(size_t)n * 64 + t] = s + bias[t];
    }
  }
}

__global__ void spgat_pool(const float* __restrict__ h, const int* __restrict__ batch,
                           float* __restrict__ sums, float* __restrict__ cnts, int Nn) {
  int i = blockIdx.x * blockDim.x + threadIdx.x;
  if (i >= Nn * 64) return;
  int n = i >> 6, c = i & 63;
  int g = batch[n];
  atomicAdd(&sums[g * 64 + c], h[(size_t)n * 64 + c]);
  if (c == 0) atomicAdd(&cnts[g], 1.0f);
}

// tiny MLP head: relu(g@M0_W + M0_b) @ M1_W + M1_b, one block per graph
__global__ __launch_bounds__(64) void spgat_mlp(
    const float* __restrict__ sums, const float* __restrict__ cnts,
    const float* __restrict__ W0, const float* __restrict__ b0,
    const float* __restrict__ W1, const float* __restrict__ b1,
    float* __restrict__ out) {
  __shared__ float g[64];
  __shared__ float t[64];
  int gi = blockIdx.x, tid = threadIdx.x;
  float c = cnts[gi]; c = c < 1.f ? 1.f : c;
  g[tid] = sums[gi * 64 + tid] / c;
  __syncthreads();
  float a = b0[tid];
  for (int k = 0; k < 64; ++k) a += g[k] * W0[k * 64 + tid];
  t[tid] = a > 0.f ? a : 0.f;
  __syncthreads();
  if (tid < 10) {
    float o = b1[tid];
    for (int k = 0; k < 64; ++k) o += t[k] * W1[k * 10 + tid];
    out[gi * 10 + tid] = o;
  }
}

extern "C" void kernel_launch(void* const* d_in, const int* in_sizes, int n_in,
                              void* d_out, int out_size, void* d_ws, size_t ws_size,
                              hipStream_t stream) {
  (void)n_in; (void)out_size; (void)ws_size;
  const float* x     = (const float*)d_in[0];
  const int*   ei    = (const int*)d_in[1];
  const int*   batch = (const int*)d_in[2];
  const float* A_W   = (const float*)d_in[3];
  const float* A_b   = (const float*)d_in[4];
  const float* W0    = (const float*)d_in[5];
  const float* as0   = (const float*)d_in[6];
  const float* ad0   = (const float*)d_in[7];
  const float* b0    = (const float*)d_in[8];
  const float* W1    = (const float*)d_in[9];
  const float* as1   = (const float*)d_in[10];
  const float* ad1   = (const float*)d_in[11];
  const float* b1    = (const float*)d_in[12];
  const float* W2    = (const float*)d_in[13];
  const float* as2   = (const float*)d_in[14];
  const float* ad2   = (const float*)d_in[15];
  const float* b2    = (const float*)d_in[16];
  const float* M0W   = (const float*)d_in[17];
  const float* M0b   = (const float*)d_in[18];
  const float* M1W   = (const float*)d_in[19];
  const float* M1b   = (const float*)d_in[20];
  float* out = (float*)d_out;

  const int Nn = in_sizes[0] / 16;   // 50000 (d_in=16)
  const int E  = in_sizes[1] / 2;    // 800000
  const int ET = E + Nn;             // logical edges incl. self loops

  // workspace carve-out
  float* ws = (float*)d_ws;
  const size_t NB = (size_t)Nn * 256;
  float* bufA  = ws;                        // activations ping
  float* bufB  = ws + NB;                   // activations pong
  float* bufW  = ws + 2 * NB;               // h = x @ W (per layer)
  float* ss    = ws + 3 * NB;               // [Nn,4] src scores
  float* sd    = ss + (size_t)Nn * 4;
  float* psum  = sd + (size_t)Nn * 4;       // [64,64] pooled sums
  float* pcnt  = psum + 64 * 64;            // [64] counts
  int*   cnt    = (int*)(pcnt + 64);        // [Nn]
  int*   start  = cnt + Nn;                 // [Nn+1]
  int*   cursor = start + Nn + 1;           // [Nn]
  int*   srcl   = cursor + Nn;              // [E+Nn] CSR source ids

  const dim3 gblk(128);
  const int mtiles  = Nn / 16;              // 3125, exact
  const int eblocks = (ET + 255) / 256;
  const int nh4b    = (Nn * 4 + 255) / 256;

  // ---- CSR build (once) ----
  spgat_ifill<<<(Nn + 255) / 256, 256, 0, stream>>>(cnt, 0, Nn);
  spgat_count<<<eblocks, 256, 0, stream>>>(ei, E, Nn, cnt);
  spgat_scan<<<1, 1024, 0, stream>>>(cnt, start, cursor, Nn, ET);
  spgat_scatter<<<eblocks, 256, 0, stream>>>(ei, E, Nn, cursor, srcl);

  // ---- h0 = x @ A_W + A_b -> bufA [Nn,64] ----
  spgat_gemm_wmma<<<dim3(mtiles, 1), gblk, 0, stream>>>(x, A_W, A_b, bufA, Nn, 64, 16);

  struct Layer { const float *W, *as_, *ad_, *b; };
  const Layer L[3] = {{W0, as0, ad0, b0}, {W1, as1, ad1, b1}, {W2, as2, ad2, b2}};
  const float* hin = bufA;
  int Kin = 64;
  float* houts[3] = {bufB, bufA, bufB};

  for (int l = 0; l < 3; ++l) {
    spgat_gemm_wmma<<<dim3(mtiles, 4), gblk, 0, stream>>>(hin, L[l].W, nullptr, bufW,
                                                          Nn, 256, Kin);
    spgat_scores<<<nh4b, 256, 0, stream>>>(bufW, L[l].as_, L[l].ad_, ss, sd, Nn);
    spgat_gather<<<Nn, 256, 0, stream>>>(srcl, start, ss, sd, bufW, L[l].b,
                                         houts[l], (l < 2) ? 0 : 1);
    if (l < 2) { hin = houts[l]; Kin = 256; }
  }

  spgat_fill<<<(64 * 64 + 64 + 255) / 256, 256, 0, stream>>>(psum, 0.f, 64 * 64 + 64);
  spgat_pool<<<(Nn * 64 + 255) / 256, 256, 0, stream>>>(bufB, batch, psum, pcnt, Nn);
  spgat_mlp<<<64, 64, 0, stream>>>(psum, pcnt, M0W, M0b, M1W, M1b, out);
}